// SNR_36249523978623
// MI455X (gfx1250) — compile-verified
//
#include <hip/hip_runtime.h>
#include <hip/hip_bf16.h>

typedef __attribute__((ext_vector_type(2))) float v2f;
typedef __attribute__((ext_vector_type(8))) float v8f;

// ---------------- degree / normalization ----------------

__global__ void SNR_deg_init(float* __restrict__ deg, int n) {
    int i = blockIdx.x * blockDim.x + threadIdx.x;
    if (i < n) deg[i] = 1.0f;   // self-loop contributes 1 to every target degree
}

__global__ void SNR_deg_acc(const int* __restrict__ col, float* __restrict__ deg, int e) {
    int i = blockIdx.x * blockDim.x + threadIdx.x;
    if (i < e) atomicAdd(&deg[col[i]], 1.0f);
}

__global__ void SNR_dis(float* __restrict__ deg, int n) {
    int i = blockIdx.x * blockDim.x + threadIdx.x;
    if (i < n) deg[i] = rsqrtf(deg[i]);   // deg >= 1 always (self loop)
}

// ---------------- GEMM1: xw = x @ W_gcn  (fp32 WMMA 16x16x4) ----------------

__global__ __launch_bounds__(256) void SNR_gemm1(const float* __restrict__ x,
                                                 const float* __restrict__ W,
                                                 float* __restrict__ xw) {
    __shared__ float xs[16 * 132];           // 16 rows of x, stride 132 to kill bank conflicts
    const int m0  = blockIdx.x * 16;
    const int tid = threadIdx.x;

    // cooperative load: 16x128 floats = 512 float4, 256 threads -> 2 each
    const float4* xg = (const float4*)(x + (size_t)m0 * 128);
    for (int i = tid; i < 16 * 32; i += 256) {
        int r = i >> 5, cc = i & 31;
        float4 v = xg[r * 32 + cc];
        float* dst = &xs[r * 132 + cc * 4];
        dst[0] = v.x; dst[1] = v.y; dst[2] = v.z; dst[3] = v.w;
    }
    __syncthreads();

    const int wave = tid >> 5;               // 0..7 -> output column tile
    const int lane = tid & 31;
    const int half = lane >> 4;              // 0/1 (K split for A/B fragments)
    const int lm   = lane & 15;              // M index (A) / N index (B,C)
    const int n0   = wave * 16;

    v8f c = {0.f, 0.f, 0.f, 0.f, 0.f, 0.f, 0.f, 0.f};
    #pragma unroll 8
    for (int kt = 0; kt < 32; ++kt) {
        const int kb = kt * 4 + half * 2;
        v2f a, b;
        a.x = xs[lm * 132 + kb];
        a.y = xs[lm * 132 + kb + 1];
        b.x = W[(size_t)(kb    ) * 128 + n0 + lm];
        b.y = W[(size_t)(kb + 1) * 128 + n0 + lm];
        c = __builtin_amdgcn_wmma_f32_16x16x4_f32(false, a, false, b, (short)0, c, false, false);
    }

    float* outp = xw + (size_t)m0 * 128 + n0 + lm;
    #pragma unroll
    for (int v = 0; v < 8; ++v)
        outp[(size_t)(v + 8 * half) * 128] = c[v];
}

// ---------------- h init: self-loop message + bias ----------------

__global__ void SNR_h_init(const float* __restrict__ xw, const float* __restrict__ dis,
                           const float* __restrict__ bg, float* __restrict__ h, int n) {
    int i = blockIdx.x * blockDim.x + threadIdx.x;   // one thread per (node, 4 feats)
    if (i < n * 32) {
        int node = i >> 5, f4 = i & 31;
        float s = dis[node]; s = s * s;
        float4 v = ((const float4*)xw)[(size_t)node * 32 + f4];
        float4 b = ((const float4*)bg)[f4];
        float4 o = { v.x * s + b.x, v.y * s + b.y, v.z * s + b.z, v.w * s + b.w };
        ((float4*)h)[(size_t)node * 32 + f4] = o;
    }
}

// ---------------- edge scatter: h[col] += norm * xw[row] ----------------

__global__ __launch_bounds__(256) void SNR_scatter(const int* __restrict__ row,
                                                   const int* __restrict__ col,
                                                   const float* __restrict__ dis,
                                                   const float* __restrict__ xw,
                                                   float* __restrict__ h, int E) {
    int gw   = (blockIdx.x * blockDim.x + threadIdx.x) >> 5;   // one wave32 per edge
    int lane = threadIdx.x & 31;
    if (gw >= E) return;
    int r = row[gw], c = col[gw];
    float nrm = dis[r] * dis[c];
    float4 v = ((const float4*)(xw + (size_t)r * 128))[lane];
    float* dst = h + (size_t)c * 128 + lane * 4;
    atomicAdd(dst + 0, v.x * nrm);
    atomicAdd(dst + 1, v.y * nrm);
    atomicAdd(dst + 2, v.z * nrm);
    atomicAdd(dst + 3, v.w * nrm);
}

// ---------------- GEMM2: out = h @ W_fc^T + b_fc  (fp32 WMMA) ----------------

__global__ __launch_bounds__(256) void SNR_gemm2(const float* __restrict__ h,
                                                 const float* __restrict__ Wfc,
                                                 const float* __restrict__ bfc,
                                                 float* __restrict__ out) {
    __shared__ float hs[32 * 132];
    const int m0  = blockIdx.x * 32;
    const int tid = threadIdx.x;

    const float4* hg = (const float4*)(h + (size_t)m0 * 128);
    for (int i = tid; i < 32 * 32; i += 256) {
        int r = i >> 5, cc = i & 31;
        float4 v = hg[r * 32 + cc];
        float* dst = &hs[r * 132 + cc * 4];
        dst[0] = v.x; dst[1] = v.y; dst[2] = v.z; dst[3] = v.w;
    }
    __syncthreads();

    const int wave = tid >> 5;
    const int lane = tid & 31;
    const int half = lane >> 4;
    const int lm   = lane & 15;
    const int mrow = (wave >> 2) * 16;     // local m tile: 0 or 16
    const int n0   = (wave & 3) * 16;      // 0..48 (64 output cols)

    float bias = bfc[n0 + lm];
    v8f c = {bias, bias, bias, bias, bias, bias, bias, bias};

    #pragma unroll 8
    for (int kt = 0; kt < 32; ++kt) {
        const int kb = kt * 4 + half * 2;
        v2f a, b;
        a.x = hs[(mrow + lm) * 132 + kb];
        a.y = hs[(mrow + lm) * 132 + kb + 1];
        // B[k][n] = W_fc[n][k]  (out = h @ W_fc^T)
        b.x = Wfc[(size_t)(n0 + lm) * 128 + kb];
        b.y = Wfc[(size_t)(n0 + lm) * 128 + kb + 1];
        c = __builtin_amdgcn_wmma_f32_16x16x4_f32(false, a, false, b, (short)0, c, false, false);
    }

    float* outp = out + (size_t)(m0 + mrow) * 64 + n0 + lm;
    #pragma unroll
    for (int v = 0; v < 8; ++v)
        outp[(size_t)(v + 8 * half) * 64] = c[v];
}

// ---------------- launch ----------------

extern "C" void kernel_launch(void* const* d_in, const int* in_sizes, int n_in,
                              void* d_out, int out_size, void* d_ws, size_t ws_size,
                              hipStream_t stream) {
    const float* x   = (const float*)d_in[0];
    const int*   ei  = (const int*)d_in[1];
    const float* Wg  = (const float*)d_in[2];
    const float* bg  = (const float*)d_in[3];
    const float* Wfc = (const float*)d_in[4];
    const float* bfc = (const float*)d_in[5];
    float* out = (float*)d_out;

    const int N = in_sizes[0] / 128;   // 100000
    const int E = in_sizes[1] / 2;     // 1600000
    const int* row = ei;               // edge_index[0] = source
    const int* col = ei + E;           // edge_index[1] = target

    char* ws = (char*)d_ws;
    float* deg = (float*)ws;                                   // N floats (becomes deg^-1/2)
    float* xw  = (float*)(ws + (1 << 20));                     // N*128 floats
    float* h   = (float*)(ws + (1 << 20) + (size_t)N * 128 * 4);

    SNR_deg_init<<<(N + 255) / 256, 256, 0, stream>>>(deg, N);
    SNR_deg_acc <<<(E + 255) / 256, 256, 0, stream>>>(col, deg, E);
    SNR_gemm1   <<<N / 16, 256, 0, stream>>>(x, Wg, xw);
    SNR_dis     <<<(N + 255) / 256, 256, 0, stream>>>(deg, N);
    SNR_h_init  <<<(N * 32 + 255) / 256, 256, 0, stream>>>(xw, deg, bg, h, N);
    SNR_scatter <<<((size_t)E * 32 + 255) / 256, 256, 0, stream>>>(row, col, deg, xw, h, E);
    SNR_gemm2   <<<N / 32, 256, 0, stream>>>(h, Wfc, bfc, out);
}